// Net_32770600469079
// MI455X (gfx1250) — compile-verified
//
#include <hip/hip_runtime.h>
#include <hip/hip_bf16.h>

typedef __attribute__((ext_vector_type(2))) float v2f;
typedef __attribute__((ext_vector_type(8))) float v8f;

#define NG 256   // NUM_GRAPHS

// ---------------------------------------------------------------------------
// 1) degree: deg[dst] += 1 over real edges (self loop added later as +1)
// ---------------------------------------------------------------------------
__global__ void k_deg(const long long* __restrict__ dst, float* __restrict__ deg, int E) {
    int e = blockIdx.x * blockDim.x + threadIdx.x;
    if (e < E) atomicAdd(&deg[(int)dst[e]], 1.0f);
}

// dinv = rsqrt(deg + 1)  (in place; +1 is the self loop, so deg>0 always)
__global__ void k_dinv(float* __restrict__ deg, int N) {
    int n = blockIdx.x * blockDim.x + threadIdx.x;
    if (n < N) deg[n] = rsqrtf(deg[n] + 1.0f);
}

// ---------------------------------------------------------------------------
// 2) p1 = (diag(dinv) @ x @ W1) via V_WMMA_F32_16X16X4_F32, 16-node tile/wave.
//    dinv folded into A (row scaling). Full tiles only -> no guards, EXEC all-1.
//    A (16x4 f32): lanes 0-15 hold {K0,K1} of row M=lane; lanes 16-31 {K2,K3}.
//    B (4x16 f32): vgpr0 = rows {K0 | K2} across lane halves, vgpr1 = {K1 | K3}.
//    D (16x16 f32): vgpr r = row (r + 8*laneHalf), col = lane%16.
// ---------------------------------------------------------------------------
__global__ void __launch_bounds__(256)
k_transform1(const float* __restrict__ x, const float* __restrict__ W1,
             const float* __restrict__ dinv, float* __restrict__ p1,
             int n_tiles) {
    int wv = blockIdx.x * (blockDim.x >> 5) + (threadIdx.x >> 5);
    if (wv >= n_tiles) return;                 // uniform per wave: EXEC stays all-1s
    int lane = threadIdx.x & 31;
    int half = lane >> 4;                      // 0: lanes 0-15, 1: lanes 16-31
    int l16  = lane & 15;

    int rowA = wv * 16 + l16;
    float di = dinv[rowA];

    v2f a, b;
    a.x = x[rowA * 4 + half * 2 + 0] * di;     // row-scale A by dinv[row]
    a.y = x[rowA * 4 + half * 2 + 1] * di;
    b.x = W1[(half * 2 + 0) * 16 + l16];
    b.y = W1[(half * 2 + 1) * 16 + l16];

    v8f c = {};
    c = __builtin_amdgcn_wmma_f32_16x16x4_f32(
            /*neg_a=*/false, a, /*neg_b=*/false, b,
            /*c_mod=*/(short)0, c, /*reuse_a=*/false, /*reuse_b=*/false);

    int rowBase = wv * 16 + half * 8;
    float* out = p1 + (size_t)rowBase * 16 + l16;
#pragma unroll
    for (int r = 0; r < 8; ++r)
        out[r * 16] = c[r];                    // 128B-coalesced store per vgpr
}

// scalar tail for N % 16 nodes (launched only if needed)
__global__ void k_transform1_tail(const float* __restrict__ x, const float* __restrict__ W1,
                                  const float* __restrict__ dinv, float* __restrict__ p1,
                                  int start, int N) {
    int n = start + blockIdx.x * blockDim.x + threadIdx.x;
    if (n >= N) return;
    float di = dinv[n];
    float x0 = x[n * 4 + 0] * di, x1 = x[n * 4 + 1] * di;
    float x2 = x[n * 4 + 2] * di, x3 = x[n * 4 + 3] * di;
#pragma unroll
    for (int j = 0; j < 16; ++j)
        p1[(size_t)n * 16 + j] =
            fmaf(x0, W1[j], fmaf(x1, W1[16 + j], fmaf(x2, W1[32 + j], x3 * W1[48 + j])));
}

// ---------------------------------------------------------------------------
// 3) edge scatter, layer 1: a1[dst][0:16] += p1[src][0:16]
// ---------------------------------------------------------------------------
__global__ void k_scatter1(const long long* __restrict__ src, const long long* __restrict__ dst,
                           const float* __restrict__ p1, float* __restrict__ a1, int E) {
    int e = blockIdx.x * blockDim.x + threadIdx.x;
    if (e >= E) return;
    int s = (int)src[e], d = (int)dst[e];
    const float4* ps = (const float4*)(p1 + (size_t)s * 16);
    float4 v0 = ps[0], v1 = ps[1], v2 = ps[2], v3 = ps[3];
    float* pd = a1 + (size_t)d * 16;
    atomicAdd(pd + 0,  v0.x); atomicAdd(pd + 1,  v0.y); atomicAdd(pd + 2,  v0.z); atomicAdd(pd + 3,  v0.w);
    atomicAdd(pd + 4,  v1.x); atomicAdd(pd + 5,  v1.y); atomicAdd(pd + 6,  v1.z); atomicAdd(pd + 7,  v1.w);
    atomicAdd(pd + 8,  v2.x); atomicAdd(pd + 9,  v2.y); atomicAdd(pd + 10, v2.z); atomicAdd(pd + 11, v2.w);
    atomicAdd(pd + 12, v3.x); atomicAdd(pd + 13, v3.y); atomicAdd(pd + 14, v3.z); atomicAdd(pd + 15, v3.w);
}

// ---------------------------------------------------------------------------
// 4) h1 = relu(dinv*(a1 + p1(self)) + b1);  p2 = (h1 @ W2) * dinv   [N,2]
// ---------------------------------------------------------------------------
__global__ void k_mid(const float* __restrict__ a1, const float* __restrict__ p1,
                      const float* __restrict__ dinv, const float* __restrict__ b1,
                      const float* __restrict__ W2, float* __restrict__ p2, int N) {
    int n = blockIdx.x * blockDim.x + threadIdx.x;
    if (n >= N) return;
    float di = dinv[n];
    float o0 = 0.0f, o1 = 0.0f;
#pragma unroll
    for (int j = 0; j < 16; ++j) {
        float v = di * (a1[(size_t)n * 16 + j] + p1[(size_t)n * 16 + j]) + b1[j];
        v = v > 0.0f ? v : 0.0f;               // ReLU
        o0 = fmaf(v, W2[j * 2 + 0], o0);
        o1 = fmaf(v, W2[j * 2 + 1], o1);
    }
    p2[(size_t)n * 2 + 0] = o0 * di;
    p2[(size_t)n * 2 + 1] = o1 * di;
}

// ---------------------------------------------------------------------------
// 5) edge scatter, layer 2: a2[dst][0:2] += p2[src][0:2]
// ---------------------------------------------------------------------------
__global__ void k_scatter2(const long long* __restrict__ src, const long long* __restrict__ dst,
                           const float* __restrict__ p2, float* __restrict__ a2, int E) {
    int e = blockIdx.x * blockDim.x + threadIdx.x;
    if (e >= E) return;
    int s = (int)src[e], d = (int)dst[e];
    float2 v = *(const float2*)(p2 + (size_t)s * 2);
    atomicAdd(a2 + (size_t)d * 2 + 0, v.x);
    atomicAdd(a2 + (size_t)d * 2 + 1, v.y);
}

// ---------------------------------------------------------------------------
// 6) head + mean-pool: h2 = dinv*(a2+p2)+b2; o = h2@Wl+bl; LDS-reduce by graph
// ---------------------------------------------------------------------------
__global__ void __launch_bounds__(256)
k_pool(const float* __restrict__ a2, const float* __restrict__ p2,
       const float* __restrict__ dinv, const float* __restrict__ b2,
       const float* __restrict__ Wl, const float* __restrict__ bl,
       const long long* __restrict__ batch,
       float* __restrict__ sums, float* __restrict__ cnt, int N) {
    __shared__ float ls[NG * 2];
    __shared__ float lc[NG];
    for (int i = threadIdx.x; i < NG * 2; i += blockDim.x) ls[i] = 0.0f;
    for (int i = threadIdx.x; i < NG;     i += blockDim.x) lc[i] = 0.0f;
    __syncthreads();

    int n = blockIdx.x * blockDim.x + threadIdx.x;
    if (n < N) {
        float di = dinv[n];
        float h0 = di * (a2[(size_t)n * 2 + 0] + p2[(size_t)n * 2 + 0]) + b2[0];
        float h1 = di * (a2[(size_t)n * 2 + 1] + p2[(size_t)n * 2 + 1]) + b2[1];
        float o0 = fmaf(h0, Wl[0], fmaf(h1, Wl[2], bl[0]));
        float o1 = fmaf(h0, Wl[1], fmaf(h1, Wl[3], bl[1]));
        int g = (int)batch[n];
        atomicAdd(&ls[g * 2 + 0], o0);
        atomicAdd(&ls[g * 2 + 1], o1);
        atomicAdd(&lc[g], 1.0f);
    }
    __syncthreads();
    for (int g = threadIdx.x; g < NG; g += blockDim.x) {
        float c = lc[g];
        if (c != 0.0f) {
            atomicAdd(&sums[g * 2 + 0], ls[g * 2 + 0]);
            atomicAdd(&sums[g * 2 + 1], ls[g * 2 + 1]);
            atomicAdd(&cnt[g], c);
        }
    }
}

// 7) out = sums / max(cnt, 1)
__global__ void k_final(const float* __restrict__ sums, const float* __restrict__ cnt,
                        float* __restrict__ out, int G) {
    int g = blockIdx.x * blockDim.x + threadIdx.x;
    if (g >= G) return;
    float c = cnt[g]; c = c > 1.0f ? c : 1.0f;
    out[g * 2 + 0] = sums[g * 2 + 0] / c;
    out[g * 2 + 1] = sums[g * 2 + 1] / c;
}

// ---------------------------------------------------------------------------
extern "C" void kernel_launch(void* const* d_in, const int* in_sizes, int n_in,
                              void* d_out, int out_size, void* d_ws, size_t ws_size,
                              hipStream_t stream) {
    const float*      x     = (const float*)d_in[0];
    const long long*  ei    = (const long long*)d_in[1];
    const long long*  batch = (const long long*)d_in[2];
    const float*      W1    = (const float*)d_in[3];
    const float*      b1    = (const float*)d_in[4];
    const float*      W2    = (const float*)d_in[5];
    const float*      b2    = (const float*)d_in[6];
    const float*      Wl    = (const float*)d_in[7];
    const float*      bl    = (const float*)d_in[8];

    const int N = in_sizes[0] / 4;           // 200000
    const int E = in_sizes[1] / 2;           // 6.4M
    const int G = out_size / 2;              // 256
    const long long* src = ei;
    const long long* dst = ei + (size_t)E;

    // workspace layout (floats): dinv | p1[N*16] | a1[N*16] | p2[N*2] | a2[N*2] | sums[512] | cnt[256]
    float* ws   = (float*)d_ws;
    float* dinv = ws;
    float* p1   = dinv + (size_t)N;
    float* a1   = p1   + (size_t)N * 16;
    float* p2   = a1   + (size_t)N * 16;
    float* a2   = p2   + (size_t)N * 2;
    float* sums = a2   + (size_t)N * 2;
    float* cnt  = sums + 2 * NG;

    // zero accumulators (capture-safe stream memsets)
    hipMemsetAsync(dinv, 0, (size_t)N * sizeof(float), stream);
    hipMemsetAsync(a1,   0, (size_t)N * 16 * sizeof(float), stream);
    hipMemsetAsync(a2,   0, (size_t)N * 2  * sizeof(float), stream);
    hipMemsetAsync(sums, 0, (size_t)(2 * NG + NG) * sizeof(float), stream);

    const int TB = 256;
    int eb = (E + TB - 1) / TB;
    int nb = (N + TB - 1) / TB;
    int n_full_tiles = N / 16;               // WMMA covers these completely
    int tb = (n_full_tiles + (TB / 32) - 1) / (TB / 32);

    k_deg       <<<eb, TB, 0, stream>>>(dst, dinv, E);
    k_dinv      <<<nb, TB, 0, stream>>>(dinv, N);
    if (n_full_tiles > 0)
        k_transform1<<<tb, TB, 0, stream>>>(x, W1, dinv, p1, n_full_tiles);
    int rem = N - n_full_tiles * 16;
    if (rem > 0)
        k_transform1_tail<<<1, TB, 0, stream>>>(x, W1, dinv, p1, n_full_tiles * 16, N);
    k_scatter1  <<<eb, TB, 0, stream>>>(src, dst, p1, a1, E);
    k_mid       <<<nb, TB, 0, stream>>>(a1, p1, dinv, b1, W2, p2, N);
    k_scatter2  <<<eb, TB, 0, stream>>>(src, dst, p2, a2, E);
    k_pool      <<<nb, TB, 0, stream>>>(a2, p2, dinv, b2, Wl, bl, batch, sums, cnt, N);
    k_final     <<<(G + TB - 1) / TB, TB, 0, stream>>>(sums, cnt, (float*)d_out, G);
}